// HMM_52733608460827
// MI455X (gfx1250) — compile-verified
//
#include <hip/hip_runtime.h>
#include <hip/hip_bf16.h>

// HMM forward-backward posterior, chunked associative scan.
//   S=65 states padded to SP=80 (5x16 WMMA tiles), K padded to 68.
//   Phase 1: per-chunk linear-space transfer matrices via V_WMMA_F32_16X16X4_F32.
//   Phase 2: serial scan of 1024 chunk operators -> entry/exit vectors + logZ.
//   Phase 3: per-chunk vector recursions (max-shifted linear) -> posterior.

typedef float v2f __attribute__((ext_vector_type(2)));
typedef float v8f __attribute__((ext_vector_type(8)));

#define S      65
#define SP     80
#define STL    81          // LDS row stride (bank-conflict-free, gcd(17,64)=1)
#define KP     68          // K padded to multiple of 4
#define NOB    256
#define TLEN   262144
#define CL     256         // chunk length
#define NC     1024        // number of chunks  (NC*CL == TLEN)

// workspace layout (float offsets)
#define OFF_TT     0                        // TtLin[80][80]  Tt[j][i]=exp(log_trans[i][j])
#define OFF_EMAX   (OFF_TT + SP*SP)         // emax[256]
#define OFF_EMT    (OFF_EMAX + NOB)         // emLinT[256][80] normalized linear emissions
#define OFF_SCALE  (OFF_EMT + NOB*SP)       // chunkScale[NC]
#define OFF_FENT   (OFF_SCALE + NC)         // fEntryLog[NC][80]
#define OFF_BEXI   (OFF_FENT + NC*SP)       // bExitLog[NC][80]
#define OFF_LOGZ   (OFF_BEXI + NC*SP)       // logZ (padded to 64)
#define OFF_CHM    (OFF_LOGZ + 64)          // chunkM[NC][80][80]

// ---------------------------------------------------------------- phase 0
__global__ __launch_bounds__(256) void hmm_prep(const float* __restrict__ lt,
                                                const float* __restrict__ lem,
                                                float* __restrict__ ws) {
  int tid = threadIdx.x;
  for (int o = tid; o < NOB; o += 256) {
    float m = -3.0e38f;
    for (int s = 0; s < S; ++s) m = fmaxf(m, lem[s * NOB + o]);
    ws[OFF_EMAX + o] = m;
    for (int s = 0; s < SP; ++s)
      ws[OFF_EMT + o * SP + s] = (s < S) ? expf(lem[s * NOB + o] - m) : 0.0f;
  }
  for (int idx = tid; idx < SP * SP; idx += 256) {
    int j = idx / SP, i = idx % SP;
    ws[OFF_TT + idx] = (i < S && j < S) ? expf(lt[i * S + j]) : 0.0f;
  }
}

// ---------------------------------------------------------------- phase 1
// One workgroup per chunk; 25 waves, each owning one 16x16 tile of the
// running 80x80 product  M <- diag(e_t) * (Tt x M), double-buffered in LDS.
__global__ __launch_bounds__(800) void hmm_chunk_mats(const int* __restrict__ obvs,
                                                      float* __restrict__ ws) {
  __shared__ float Tt[SP * STL];
  __shared__ float Mb[2][SP * STL];
  __shared__ float ev[SP];
  __shared__ float red[800];
  __shared__ float sc[2];

  const int tid = threadIdx.x;
  const int c   = blockIdx.x;
  const float* gTt  = ws + OFF_TT;
  const float* emax = ws + OFF_EMAX;
  const float* emt  = ws + OFF_EMT;

  for (int idx = tid; idx < SP * SP; idx += 800) {
    int r = idx / SP, q = idx % SP;
    Tt[r * STL + q]    = gTt[idx];
    Mb[0][r * STL + q] = (r == q && r < S) ? 1.0f : 0.0f;
    Mb[1][r * STL + q] = 0.0f;
  }
  if (tid == 0) sc[0] = 0.0f;
  __syncthreads();

  const int lane = tid & 31;
  const int wv   = tid >> 5;          // 0..24
  const int ti   = wv / 5, tj = wv % 5;
  const int half = lane >> 4, lr = lane & 15;
  const int arow = ti * 16 + lr;      // output row of this lane's A fragment
  const int bcol = tj * 16 + lr;      // output col of this lane's B/C fragment
  const int base = c * CL;

  for (int ts = 0; ts < CL; ++ts) {
    int obs = obvs[base + ts];
    if (tid < SP)  ev[tid] = emt[obs * SP + tid];   // zero in padded rows
    if (tid == 0)  sc[0] += emax[obs];
    __syncthreads();

    const int cur = ts & 1, nxt = cur ^ 1;
    const float* Min = Mb[cur];
    v8f acc = {};
    for (int k = 0; k < KP; k += 4) {
      // A (16x4 f32): lanes 0-15 -> K = k,k+1 ; lanes 16-31 -> K = k+2,k+3
      v2f a, b;
      a[0] = Tt[arow * STL + k + 2 * half];
      a[1] = Tt[arow * STL + k + 2 * half + 1];
      // B (4x16 f32): VGPR0 rows k / k+2, VGPR1 rows k+1 / k+3 (C-layout analog)
      b[0] = Min[(k + 2 * half) * STL + bcol];
      b[1] = Min[(k + 2 * half + 1) * STL + bcol];
      acc = __builtin_amdgcn_wmma_f32_16x16x4_f32(false, a, false, b,
                                                  (short)0, acc, false, false);
    }
    float* Mo = Mb[nxt];
#pragma unroll
    for (int r = 0; r < 8; ++r) {
      int row = ti * 16 + r + half * 8;             // C layout: VGPR r -> M=r / r+8
      Mo[row * STL + bcol] = acc[r] * ev[row];
    }
    __syncthreads();

    if ((ts & 7) == 7) {                            // renormalize by max, keep f32 range
      float m = 0.0f;
      for (int idx = tid; idx < SP * SP; idx += 800)
        m = fmaxf(m, Mo[(idx / SP) * STL + (idx % SP)]);
      red[tid] = m;
      __syncthreads();
      if (tid < 32) {
        float mm = red[tid];
        for (int j = tid + 32; j < 800; j += 32) mm = fmaxf(mm, red[j]);
        red[tid] = mm;
      }
      __syncthreads();
      if (tid == 0) {
        float mm = red[0];
        for (int j = 1; j < 32; ++j) mm = fmaxf(mm, red[j]);
        mm = fmaxf(mm, 1.0e-30f);
        sc[1] = mm;
        sc[0] += logf(mm);
      }
      __syncthreads();
      float inv = 1.0f / sc[1];
      for (int idx = tid; idx < SP * SP; idx += 800)
        Mo[(idx / SP) * STL + (idx % SP)] *= inv;
      __syncthreads();
    }
  }

  // CL is even -> final matrix sits in buffer 0
  float* gM = ws + OFF_CHM + (size_t)c * SP * SP;
  for (int idx = tid; idx < SP * SP; idx += 800)
    gM[idx] = Mb[0][(idx / SP) * STL + (idx % SP)];
  if (tid == 0) ws[OFF_SCALE + c] = sc[0];
}

// ---------------------------------------------------------------- phase 2
__global__ __launch_bounds__(128) void hmm_scan(const float* __restrict__ lt,
                                                float* __restrict__ ws) {
  __shared__ float lv[SP], nv[SP], red[128];
  const int tid = threadIdx.x;
  const float* chM = ws + OFF_CHM;
  const float* csc = ws + OFF_SCALE;

  // ---- forward entry vectors: v_0 = log pi, v_{c+1} = log(F_c exp(v_c)) + scale_c
  if (tid < SP) lv[tid] = (tid == 0) ? 0.0f : ((tid < S) ? -99.0f : -3.0e38f);
  __syncthreads();
  for (int c = 0; c < NC; ++c) {
    if (tid < SP) ws[OFF_FENT + c * SP + tid] = lv[tid];
    red[tid] = (tid < S) ? lv[tid] : -3.0e38f;
    __syncthreads();
    for (int s = 64; s > 0; s >>= 1) { if (tid < s) red[tid] = fmaxf(red[tid], red[tid + s]); __syncthreads(); }
    float m = red[0];
    __syncthreads();
    if (tid < SP) nv[tid] = (tid < S) ? expf(lv[tid] - m) : 0.0f;
    __syncthreads();
    if (tid < S) {
      const float* M = chM + (size_t)c * SP * SP + tid * SP;
      float acc = 0.0f;
      for (int i = 0; i < S; ++i) acc += M[i] * nv[i];
      lv[tid] = logf(fmaxf(acc, 1.0e-37f)) + m + csc[c];
    }
    __syncthreads();
  }

  // ---- logZ = lse(f_{T-1} + log_trans[:,0])
  red[tid] = (tid < S) ? lv[tid] + lt[tid * S] : -3.0e38f;
  __syncthreads();
  for (int s = 64; s > 0; s >>= 1) { if (tid < s) red[tid] = fmaxf(red[tid], red[tid + s]); __syncthreads(); }
  float mz = red[0];
  __syncthreads();
  red[tid] = (tid < S) ? expf(lv[tid] + lt[tid * S] - mz) : 0.0f;
  __syncthreads();
  for (int s = 64; s > 0; s >>= 1) { if (tid < s) red[tid] += red[tid + s]; __syncthreads(); }
  if (tid == 0) ws[OFF_LOGZ] = mz + logf(red[0]);
  __syncthreads();

  // ---- backward exit vectors: w_{NC-1} = log_trans[:,0], w_{c-1} = log(F_c^T exp(w_c)) + scale_c
  if (tid < SP) lv[tid] = (tid < S) ? lt[tid * S] : -3.0e38f;
  __syncthreads();
  for (int c = NC - 1; c >= 0; --c) {
    if (tid < SP) ws[OFF_BEXI + c * SP + tid] = lv[tid];
    red[tid] = (tid < S) ? lv[tid] : -3.0e38f;
    __syncthreads();
    for (int s = 64; s > 0; s >>= 1) { if (tid < s) red[tid] = fmaxf(red[tid], red[tid + s]); __syncthreads(); }
    float m = red[0];
    __syncthreads();
    if (tid < SP) nv[tid] = (tid < S) ? expf(lv[tid] - m) : 0.0f;
    __syncthreads();
    if (tid < S) {
      const float* M = chM + (size_t)c * SP * SP;
      float acc = 0.0f;
      for (int j = 0; j < S; ++j) acc += M[j * SP + tid] * nv[j];
      lv[tid] = logf(fmaxf(acc, 1.0e-37f)) + m + csc[c];
    }
    __syncthreads();
  }
}

// ---------------------------------------------------------------- phase 3
__global__ __launch_bounds__(128) void hmm_posterior(const int* __restrict__ obvs,
                                                     const float* __restrict__ lt,
                                                     const float* __restrict__ lem,
                                                     const float* __restrict__ ws,
                                                     float* __restrict__ out) {
  __shared__ float TtL[SP * STL];
  __shared__ float fLog[CL][SP];
  __shared__ float cv[SP], sl[SP], red[128];
  const int tid = threadIdx.x;
  const int c   = blockIdx.x;
  const int base = c * CL;

  for (int idx = tid; idx < SP * SP; idx += 128)
    TtL[(idx / SP) * STL + (idx % SP)] = ws[OFF_TT + idx];
  if (tid < SP) cv[tid] = ws[OFF_FENT + c * SP + tid];     // log f_{base-1}
  const float lz = ws[OFF_LOGZ];
  __syncthreads();

  // ---- forward recompute, buffer log f_t in LDS
  for (int ts = 0; ts < CL; ++ts) {
    int obs = obvs[base + ts];
    red[tid] = (tid < S) ? cv[tid] : -3.0e38f;
    __syncthreads();
    for (int s = 64; s > 0; s >>= 1) { if (tid < s) red[tid] = fmaxf(red[tid], red[tid + s]); __syncthreads(); }
    float m = red[0];
    __syncthreads();
    if (tid < SP) sl[tid] = (tid < S) ? expf(cv[tid] - m) : 0.0f;
    __syncthreads();
    if (tid < SP) {
      float v = -3.0e38f;
      if (tid < S) {
        const float* Tr = TtL + tid * STL;
        float acc = 0.0f;
        for (int i = 0; i < S; ++i) acc += Tr[i] * sl[i];
        v = lem[tid * NOB + obs] + logf(fmaxf(acc, 1.0e-37f)) + m;
      }
      cv[tid] = v;
      fLog[ts][tid] = v;
    }
    __syncthreads();
  }

  // ---- backward + posterior:  out[t] = f_t + b_{t-1} - logZ  (ref index shift)
  if (tid < SP) cv[tid] = ws[OFF_BEXI + c * SP + tid];     // log b_{base+CL-1}
  __syncthreads();
  for (int ts = CL - 1; ts >= 0; --ts) {
    int t   = base + ts;
    int obs = obvs[t];
    float sv = (tid < S) ? lem[tid * NOB + obs] + cv[tid] : -3.0e38f;
    red[tid] = sv;
    __syncthreads();
    for (int s = 64; s > 0; s >>= 1) { if (tid < s) red[tid] = fmaxf(red[tid], red[tid + s]); __syncthreads(); }
    float m = red[0];
    __syncthreads();
    if (tid < SP) sl[tid] = (tid < S) ? expf(sv - m) : 0.0f;
    __syncthreads();
    float bw = -3.0e38f;
    if (tid < S) {
      float acc = 0.0f;
      for (int j = 0; j < S; ++j) acc += TtL[j * STL + tid] * sl[j];
      bw = logf(fmaxf(acc, 1.0e-37f)) + m;                 // b_{t-1}[tid]
      if (c == 0 && ts == 0)                               // ref's special row 0
        bw = lt[tid] + lem[tid * NOB + obs] + cv[tid];
      out[(size_t)t * S + tid] = fLog[ts][tid] + bw - lz;
    }
    __syncthreads();
    if (tid < SP) cv[tid] = (tid < S) ? bw : -3.0e38f;
    __syncthreads();
  }
}

// ---------------------------------------------------------------- launch
extern "C" void kernel_launch(void* const* d_in, const int* in_sizes, int n_in,
                              void* d_out, int out_size, void* d_ws, size_t ws_size,
                              hipStream_t stream) {
  (void)in_sizes; (void)n_in; (void)out_size; (void)ws_size;
  const int*   obvs = (const int*)d_in[0];
  const float* lt   = (const float*)d_in[1];   // log_trans [65][65]
  const float* lem  = (const float*)d_in[2];   // log_em    [65][256]
  float* out = (float*)d_out;
  float* ws  = (float*)d_ws;

  hipLaunchKernelGGL(hmm_prep,       dim3(1),  dim3(256), 0, stream, lt, lem, ws);
  hipLaunchKernelGGL(hmm_chunk_mats, dim3(NC), dim3(800), 0, stream, obvs, ws);
  hipLaunchKernelGGL(hmm_scan,       dim3(1),  dim3(128), 0, stream, lt, ws);
  hipLaunchKernelGGL(hmm_posterior,  dim3(NC), dim3(128), 0, stream, obvs, lt, lem, ws, out);
}